// LQG_26654567039526
// MI455X (gfx1250) — compile-verified
//
#include <hip/hip_runtime.h>
#include <hip/hip_bf16.h>

typedef __attribute__((ext_vector_type(2))) float v2f;
typedef __attribute__((ext_vector_type(8))) float v8f;

#define NTRIALS 256
#define DD      256
#define MM      128
#define TT      512
#define SW      512      // state width: [x | xh]
#define WN      1024     // fused weight columns
#define NPART   128      // cost partial slots: 8 rowTiles x 16 quad wave-cols

// LDS swizzle: addr(r,c) = r*512 + (c ^ (4*(r&15)))  — kills the 16-way bank
// conflict of the A-fragment gather (16 rows, same k) while preserving
// pair/quad contiguity (swizzle only touches bits >= 2).
__device__ __forceinline__ int lswz(int r, int c) {
    return (r << 9) + (c ^ ((r & 15) << 2));
}

// ---------------- prologue helper GEMMs (one-time, trivial FLOPs) -------------

// Z[i*N+j] = sum_k X[i*Kk+k] * Y[k*N+j]
__global__ void gemm_nn(const float* __restrict__ X, const float* __restrict__ Y,
                        float* __restrict__ Z, int Mr, int N, int Kk) {
    int idx = blockIdx.x * blockDim.x + threadIdx.x;
    if (idx >= Mr * N) return;
    int i = idx / N, j = idx % N;
    float s = 0.f;
    for (int k = 0; k < Kk; ++k) s += X[i * Kk + k] * Y[k * N + j];
    Z[idx] = s;
}

// Z[i*N+j] = sum_k X[k*ldX+i] * Y[k*N+j]   (X transposed)
__global__ void gemm_tn(const float* __restrict__ X, const float* __restrict__ Y,
                        float* __restrict__ Z, int Mo, int N, int Kk, int ldX) {
    int idx = blockIdx.x * blockDim.x + threadIdx.x;
    if (idx >= Mo * N) return;
    int i = idx / N, j = idx % N;
    float s = 0.f;
    for (int k = 0; k < Kk; ++k) s += X[k * ldX + i] * Y[k * N + j];
    Z[idx] = s;
}

// Build fused weight Wbig (SW x WN), row-major [c][j].
//  j<256        : c<256 -> A[j][c]            ; else -> -G3[j][c-256]          (x' cols)
//  256<=j<512   : c<256 -> G1[j'][c]          ; else -> A-G3-G1 at [j'][c-256] (xh' cols)
//  512<=j<768   : c<256 -> G4[c][q]           ; else -> 0                      (x quad)
//  768<=j<1024  : c<256 -> 0                  ; else -> G5[c-256][q]           (xh quad)
__global__ void build_W(const float* __restrict__ A, const float* __restrict__ G1,
                        const float* __restrict__ G3, const float* __restrict__ G4,
                        const float* __restrict__ G5, float* __restrict__ Wbig) {
    int idx = blockIdx.x * blockDim.x + threadIdx.x;
    if (idx >= SW * WN) return;
    int c = idx / WN, j = idx % WN;
    float v;
    if (j < 256) {
        v = (c < 256) ? A[j * DD + c] : -G3[j * DD + (c - 256)];
    } else if (j < 512) {
        int jp = j - 256;
        if (c < 256) v = G1[jp * DD + c];
        else { int cc = c - 256; v = A[jp * DD + cc] - G3[jp * DD + cc] - G1[jp * DD + cc]; }
    } else if (j < 768) {
        int q = j - 512;
        v = (c < 256) ? G4[c * DD + q] : 0.f;
    } else {
        int q = j - 768;
        v = (c >= 256) ? G5[(c - 256) * DD + q] : 0.f;
    }
    Wbig[c * WN + j] = v;
}

// X0 = [x0 | 0]
__global__ void init_X(const float* __restrict__ x0, float* __restrict__ X) {
    int idx = blockIdx.x * blockDim.x + threadIdx.x;
    if (idx >= NTRIALS * SW) return;
    int n = idx / SW, c = idx % SW;
    X[idx] = (c < 256) ? x0[n * DD + c] : 0.f;
}

__global__ void final_reduce(const float* __restrict__ p, float* __restrict__ out) {
    if (blockIdx.x == 0 && threadIdx.x == 0) {
        float s = 0.f;
        for (int i = 0; i < NPART; ++i) s += p[i];
        *out = s * (1.0f / (float)NTRIALS);
    }
}

// ---------------- per-step fused kernel (f32 WMMA, 2x2 register blocking) -----
// grid (8, 8): 8 x 32-row tiles, 8 x 128-col groups; 128 thr = 4 wave32.
// Each wave owns a 32x32 output block = 2x2 WMMA tiles: per K-step of 4,
// 2 A-frag LDS b64 loads + 2 B-frag global loads feed 4 v_wmma_f32_16x16x4_f32.
// Col groups 0..3 -> new state (+ w_t); 4..7 -> quadratic-form cost partials
// written to fixed per-wave slots (atomic-free, deterministic).  Block (0,0)
// folds the reduction of the PREVIOUS step's partials into d_out[t-1].
__global__ __launch_bounds__(128) void lqg_step(
    const float* __restrict__ Xc, float* __restrict__ Xn,
    const float* __restrict__ W, const float* __restrict__ wt,
    float* __restrict__ pcur, const float* __restrict__ pprev,
    float* __restrict__ cost_prev, int reduce_prev) {
    __shared__ __align__(16) float lx[32 * 512];   // exactly 64 KB

    const int tid     = threadIdx.x;
    const int rowTile = blockIdx.x;          // 0..7 (32 rows each)
    const int j0      = blockIdx.y * 128;    // 0,128,...,896
    const int row0    = rowTile * 32;

    // warm L2/L0 with this block's w_t tile (global_prefetch_b8, no counters)
    // while the K-loop runs; only state col-groups 0..1 consume w_t.
    if (j0 < 256) {
        int pr = tid >> 2, pc = (tid & 3) << 5;          // 32 rows x 4 lines
        __builtin_prefetch(&wt[(row0 + pr) * DD + j0 + pc], 0, 3);
    }

    // stage 32 x 512 X tile into LDS (float4, swizzled)
    const float4* Xc4 = (const float4*)Xc;
    for (int i = tid; i < 32 * 128; i += 128) {
        int r = i >> 7, c = (i & 127) << 2;
        float4 v = Xc4[(row0 + r) * 128 + (c >> 2)];
        *(float4*)&lx[lswz(r, c)] = v;
    }

    // fold: reduce previous step's partials (deterministic fixed order)
    if (reduce_prev && rowTile == 0 && blockIdx.y == 0 && tid == 0) {
        float s = 0.f;
        for (int i = 0; i < NPART; ++i) s += pprev[i];
        *cost_prev = s * (1.0f / (float)NTRIALS);
    }
    __syncthreads();

    const int wave = tid >> 5;
    const int lane = tid & 31;
    const int half = lane >> 4;          // 0: lanes 0-15, 1: lanes 16-31
    const int m    = lane & 15;          // A-row within tile / D-col
    const int kb   = half * 2;           // A/B K sub-offset per ISA 16x4 layout
    const int jn0  = j0 + wave * 32 + m; // first N tile column
    const int swz  = m << 2;

    v8f a00 = {}, a01 = {}, a10 = {}, a11 = {};
    for (int k0 = 0; k0 < SW; k0 += 4) {
        const int ca = k0 + kb;
        const int cs = ca ^ swz;         // even -> 8B-aligned pair (cs, cs+1)
        // A fragments as b64 LDS loads straight into even-aligned reg pairs
        v2f A0 = *(const v2f*)&lx[(m << 9) + cs];
        v2f A1 = *(const v2f*)&lx[((m + 16) << 9) + cs];
        const float* wp = W + (size_t)ca * WN + jn0;
        v2f B0, B1;
        B0.x = wp[0];  B0.y = wp[WN];
        B1.x = wp[16]; B1.y = wp[WN + 16];
        a00 = __builtin_amdgcn_wmma_f32_16x16x4_f32(false, A0, false, B0, (short)0, a00, false, false);
        a01 = __builtin_amdgcn_wmma_f32_16x16x4_f32(false, A0, false, B1, (short)0, a01, false, false);
        a10 = __builtin_amdgcn_wmma_f32_16x16x4_f32(false, A1, false, B0, (short)0, a10, false, false);
        a11 = __builtin_amdgcn_wmma_f32_16x16x4_f32(false, A1, false, B1, (short)0, a11, false, false);
    }

    const int jn1 = jn0 + 16;
    if (j0 < SW) {
        // state epilogue: D layout VGPR r -> M = r + 8*half, N = m
        for (int r = 0; r < 8; ++r) {
            int mr   = r + (half << 3);
            int g0   = row0 + mr;        // M tile 0 global row
            int g1   = g0 + 16;          // M tile 1 global row
            float v00 = a00[r], v01 = a01[r], v10 = a10[r], v11 = a11[r];
            if (jn0 < 256) { v00 += wt[g0 * DD + jn0]; v10 += wt[g1 * DD + jn0]; }
            if (jn1 < 256) { v01 += wt[g0 * DD + jn1]; v11 += wt[g1 * DD + jn1]; }
            Xn[g0 * SW + jn0] = v00;
            Xn[g0 * SW + jn1] = v01;
            Xn[g1 * SW + jn0] = v10;
            Xn[g1 * SW + jn1] = v11;
        }
    } else {
        // quadratic-form epilogue: cost partial = sum D[m][j] * X[m][j-512]
        const int c0 = jn0 - SW, c1 = c0 + 16;
        float p = 0.f;
        for (int r = 0; r < 8; ++r) {
            int mr0 = r + (half << 3);
            int mr1 = mr0 + 16;
            p += a00[r] * lx[lswz(mr0, c0)];
            p += a01[r] * lx[lswz(mr0, c1)];
            p += a10[r] * lx[lswz(mr1, c0)];
            p += a11[r] * lx[lswz(mr1, c1)];
        }
        for (int off = 16; off > 0; off >>= 1) p += __shfl_xor(p, off, 32);
        if (lane == 0)
            pcur[rowTile * 16 + (blockIdx.y - 4) * 4 + wave] = p;  // fixed slot
    }
}

// ---------------- host-side launch -------------------------------------------

extern "C" void kernel_launch(void* const* d_in, const int* in_sizes, int n_in,
                              void* d_out, int out_size, void* d_ws, size_t ws_size,
                              hipStream_t stream) {
    const float* A  = (const float*)d_in[0];   // (256,256)
    const float* B  = (const float*)d_in[1];   // (256,128)
    const float* C  = (const float*)d_in[2];   // (256,256)
    const float* Q  = (const float*)d_in[3];   // (256,256)
    const float* R  = (const float*)d_in[4];   // (128,128)
    const float* K  = (const float*)d_in[5];   // (128,256)
    const float* L  = (const float*)d_in[6];   // (256,256)
    const float* x0 = (const float*)d_in[7];   // (256,256)
    const float* w  = (const float*)d_in[8];   // (512,256,256)
    float* out = (float*)d_out;                // (512,)

    float* ws   = (float*)d_ws;                // ~4.6 MB used
    float* Wbig = ws;                 // 512*1024
    float* Xa   = Wbig + SW * WN;     // 256*512
    float* Xb   = Xa + NTRIALS * SW;  // 256*512
    float* G1   = Xb + NTRIALS * SW;  // 256*256 : L@C
    float* G3   = G1 + DD * DD;       // 256*256 : B@K
    float* QC   = G3 + DD * DD;       // 256*256 : Q_obs@C
    float* G4   = QC + DD * DD;       // 256*256 : C^T Q C
    float* RK   = G4 + DD * DD;       // 128*256 : R@K
    float* G5   = RK + MM * DD;       // 256*256 : K^T R K
    float* pA   = G5 + DD * DD;       // NPART
    float* pB   = pA + NPART;         // NPART

    // prologue: combined matrices + fused weight + initial state
    gemm_nn<<<256, 256, 0, stream>>>(B, K, G3, 256, 256, 128);
    gemm_nn<<<256, 256, 0, stream>>>(L, C, G1, 256, 256, 256);
    gemm_nn<<<256, 256, 0, stream>>>(Q, C, QC, 256, 256, 256);
    gemm_tn<<<256, 256, 0, stream>>>(C, QC, G4, 256, 256, 256, 256);
    gemm_nn<<<128, 256, 0, stream>>>(R, K, RK, 128, 256, 128);
    gemm_tn<<<256, 256, 0, stream>>>(K, RK, G5, 256, 256, 128, 256);
    build_W<<<(SW * WN) / 256, 256, 0, stream>>>(A, G1, G3, G4, G5, Wbig);
    init_X<<<(NTRIALS * SW) / 256, 256, 0, stream>>>(x0, Xa);

    // rollout: one fused WMMA GEMM per step, cost reduction folded into t+1
    for (int t = 0; t < TT; ++t) {
        const float* Xcur = (t & 1) ? Xb : Xa;
        float*       Xnxt = (t & 1) ? Xa : Xb;
        float*       pc   = (t & 1) ? pB : pA;
        const float* pp   = (t & 1) ? pA : pB;
        lqg_step<<<dim3(8, 8), 128, 0, stream>>>(
            Xcur, Xnxt, Wbig, w + (size_t)t * NTRIALS * DD,
            pc, pp, (t > 0) ? (out + (t - 1)) : out, (t > 0) ? 1 : 0);
    }
    final_reduce<<<1, 32, 0, stream>>>(pB, out + (TT - 1));
}